// Leaf_attention_41068477284682
// MI455X (gfx1250) — compile-verified
//
#include <hip/hip_runtime.h>

// Problem constants (match reference)
#define VV   100000   // vocab / vertex count
#define DD   128      // embedding dim
#define AA   64       // attention hidden dim
#define LVLS 3        // tree levels
#define NN   20000    // nodes per level
#define KK   16       // neighbors per node

typedef float v2f __attribute__((ext_vector_type(2)));
typedef float v8f __attribute__((ext_vector_type(8)));

// D = A(16x4 f32) * B(4x16 f32) + C(16x16 f32), wave32, one tile per wave.
__device__ __forceinline__ v8f wmma_f32(v2f a, v2f b, v8f c) {
  return __builtin_amdgcn_wmma_f32_16x16x4_f32(
      /*neg_a=*/false, a, /*neg_b=*/false, b,
      /*c_mod=*/(short)0, c, /*reuse_a=*/false, /*reuse_b=*/false);
}

__device__ __forceinline__ void lds_fence() {
  // DS ops are in-order per wave; ensure results are consumable.
  asm volatile("s_wait_dscnt 0" ::: "memory");
}

// Transpose W_att [256][64] row-major -> WT [64][256] (k-contiguous per column)
__global__ __launch_bounds__(256) void transpose_watt(const float* __restrict__ W,
                                                      float* __restrict__ WT) {
  int i = blockIdx.x * 256 + threadIdx.x;      // 0 .. 16383
  int k = i >> 6;
  int c = i & 63;
  WT[c * 256 + k] = W[i];
}

// One wave per node. Computes temp_emb[n][0:128] for level `lvl`.
__global__ __launch_bounds__(128) void attn_kernel(
    int lvl,
    const float* __restrict__ Leaf,      // original Leaf_emb [V][D]
    const float* __restrict__ Wtmp,      // running embeddings [V][D]
    const int*   __restrict__ nodes,     // [L][N]
    const int*   __restrict__ neighbors, // [L][N][K]
    const float* __restrict__ masks,     // [L][N][K]
    const float* __restrict__ weights,   // [L][N][K]
    const float* __restrict__ WT,        // transposed W_att [64][256]
    const float* __restrict__ b_att,     // [64]
    const float* __restrict__ v_att,     // [64]
    float*       __restrict__ temp)      // [N][D] out
{
  __shared__ float sTile[4][16 * 132];   // neighbor rows, padded stride 132
  __shared__ float sNode[4][128];        // node row
  __shared__ int   sIdx[4][16];
  __shared__ float sPre[4][16];
  __shared__ float sAtt[4][16];

  const int lane = threadIdx.x & 31;
  const int w    = threadIdx.x >> 5;
  const int col  = lane & 15;            // N-column within tile / neighbor id
  const int kk   = (lane >> 4) * 2;      // K sub-offset for A/B frags
  const int n    = blockIdx.x * 4 + w;   // node index (N divisible by 4)

  const size_t nkbase = ((size_t)lvl * NN + (size_t)n) * KK;

  // --- gather: neighbor indices -> LDS, then 16 coalesced 512B row copies ---
  int idx = neighbors[nkbase + col];
  if (lane < 16) sIdx[w][lane] = idx;
  lds_fence();

#pragma unroll
  for (int r = 0; r < 16; r++) {
    int ridx = sIdx[w][r];
    float4 vrow = *(const float4*)(Wtmp + (size_t)ridx * DD + lane * 4);
    *(float4*)(&sTile[w][r * 132 + lane * 4]) = vrow;
  }
  {
    int nn = nodes[(size_t)lvl * NN + n];
    float4 vrow = *(const float4*)(Leaf + (size_t)nn * DD + lane * 4);
    *(float4*)(&sNode[w][lane * 4]) = vrow;
  }
  lds_fence();

  // --- GEMM: [16 x 256] @ [256 x 64] via f32 WMMA, K-concat of node|neighbor ---
  v8f acc0 = {}, acc1 = {}, acc2 = {}, acc3 = {};
  const float* nb  = sNode[w];
  const float* tb  = sTile[w] + col * 132;
  const float* wt0 = WT + (size_t)col * 256 + kk;
  const float* wt1 = wt0 + 16 * 256;
  const float* wt2 = wt0 + 32 * 256;
  const float* wt3 = wt0 + 48 * 256;

#pragma unroll 4
  for (int k0 = 0; k0 < 128; k0 += 4) {   // node half: A rows all = node_emb
    v2f a = *(const v2f*)(nb + k0 + kk);
    acc0 = wmma_f32(a, *(const v2f*)(wt0 + k0), acc0);
    acc1 = wmma_f32(a, *(const v2f*)(wt1 + k0), acc1);
    acc2 = wmma_f32(a, *(const v2f*)(wt2 + k0), acc2);
    acc3 = wmma_f32(a, *(const v2f*)(wt3 + k0), acc3);
  }
#pragma unroll 4
  for (int k0 = 0; k0 < 128; k0 += 4) {   // neighbor half
    v2f a = *(const v2f*)(tb + k0 + kk);
    acc0 = wmma_f32(a, *(const v2f*)(wt0 + 128 + k0), acc0);
    acc1 = wmma_f32(a, *(const v2f*)(wt1 + 128 + k0), acc1);
    acc2 = wmma_f32(a, *(const v2f*)(wt2 + 128 + k0), acc2);
    acc3 = wmma_f32(a, *(const v2f*)(wt3 + 128 + k0), acc3);
  }

  // --- epilogue: bias + leaky_relu + dot(v_att), rows reduced across lanes ---
  const float va0 = v_att[col], va1 = v_att[col + 16],
              va2 = v_att[col + 32], va3 = v_att[col + 48];
  const float ba0 = b_att[col], ba1 = b_att[col + 16],
              ba2 = b_att[col + 32], ba3 = b_att[col + 48];

#pragma unroll
  for (int vv = 0; vv < 8; vv++) {
    float x0 = acc0[vv] + ba0; x0 = x0 > 0.f ? x0 : 0.01f * x0;
    float x1 = acc1[vv] + ba1; x1 = x1 > 0.f ? x1 : 0.01f * x1;
    float x2 = acc2[vv] + ba2; x2 = x2 > 0.f ? x2 : 0.01f * x2;
    float x3 = acc3[vv] + ba3; x3 = x3 > 0.f ? x3 : 0.01f * x3;
    float t = x0 * va0 + x1 * va1 + x2 * va2 + x3 * va3;
    t += __shfl_xor(t, 1);
    t += __shfl_xor(t, 2);
    t += __shfl_xor(t, 4);
    t += __shfl_xor(t, 8);
    // row m = vv (lanes 0-15) or vv+8 (lanes 16-31)
    if (col == 0) sPre[w][(lane >> 4) * 8 + vv] = t;
  }
  lds_fence();

  // --- dual softmax over K=16 (lane k = col; both halves compute identically) ---
  float p   = sPre[w][col];
  float mk  = masks[nkbase + col];
  float wgt = weights[nkbase + col];

  float z = p + mk;
  float mx = z;
  mx = fmaxf(mx, __shfl_xor(mx, 1));
  mx = fmaxf(mx, __shfl_xor(mx, 2));
  mx = fmaxf(mx, __shfl_xor(mx, 4));
  mx = fmaxf(mx, __shfl_xor(mx, 8));
  float e1 = expf(z - mx);
  float s1 = e1;
  s1 += __shfl_xor(s1, 1); s1 += __shfl_xor(s1, 2);
  s1 += __shfl_xor(s1, 4); s1 += __shfl_xor(s1, 8);

  float mw = wgt;
  mw = fmaxf(mw, __shfl_xor(mw, 1));
  mw = fmaxf(mw, __shfl_xor(mw, 2));
  mw = fmaxf(mw, __shfl_xor(mw, 4));
  mw = fmaxf(mw, __shfl_xor(mw, 8));
  float e2 = expf(wgt - mw);
  float s2 = e2;
  s2 += __shfl_xor(s2, 1); s2 += __shfl_xor(s2, 2);
  s2 += __shfl_xor(s2, 4); s2 += __shfl_xor(s2, 8);

  float att = (e1 / s1) * (e2 / s2);
  if (lane < 16) sAtt[w][lane] = att;
  lds_fence();

  // --- temp_emb[d] = sum_k att[k] * nbr[k][d], each lane owns 4 columns ---
  float4 o = make_float4(0.f, 0.f, 0.f, 0.f);
#pragma unroll
  for (int r = 0; r < 16; r++) {
    float ar = sAtt[w][r];
    float4 tv = *(const float4*)(&sTile[w][r * 132 + lane * 4]);
    o.x += ar * tv.x; o.y += ar * tv.y; o.z += ar * tv.z; o.w += ar * tv.w;
  }
  *(float4*)(temp + (size_t)n * DD + lane * 4) = o;
}

// Deterministic duplicate resolution: highest n wins (last occurrence).
__global__ __launch_bounds__(256) void owner_kernel(int lvl,
                                                    const int* __restrict__ nodes,
                                                    int* __restrict__ owner) {
  int i = blockIdx.x * 256 + threadIdx.x;
  if (i < NN) atomicMax(&owner[nodes[(size_t)lvl * NN + i]], i);
}

__global__ __launch_bounds__(256) void scatter_kernel(int lvl,
                                                      const int* __restrict__ nodes,
                                                      const int* __restrict__ owner,
                                                      const float* __restrict__ temp,
                                                      float* __restrict__ Wtmp) {
  int t = blockIdx.x * 256 + threadIdx.x;     // 0 .. N*D-1
  int n = t >> 7;
  int d = t & 127;
  int nd = nodes[(size_t)lvl * NN + n];
  if (owner[nd] == n)
    Wtmp[(size_t)nd * DD + d] = temp[(size_t)n * DD + d];
}

extern "C" void kernel_launch(void* const* d_in, const int* in_sizes, int n_in,
                              void* d_out, int out_size, void* d_ws, size_t ws_size,
                              hipStream_t stream) {
  const float* Leaf      = (const float*)d_in[0];
  const int*   nodes     = (const int*)  d_in[1];
  const int*   neighbors = (const int*)  d_in[2];
  const float* masks     = (const float*)d_in[3];
  const float* weights   = (const float*)d_in[4];
  const float* W_att     = (const float*)d_in[5];
  const float* b_att     = (const float*)d_in[6];
  const float* v_att     = (const float*)d_in[7];
  float* Wtmp = (float*)d_out;

  char* ws = (char*)d_ws;
  float* WT   = (float*)ws;                                   // 64*256*4 = 64 KB
  float* temp = (float*)(ws + 65536);                         // N*D*4 = 10.24 MB
  int*  owner = (int*)  (ws + 65536 + (size_t)NN * DD * 4);   // V*4 = 400 KB

  // W_tmp := Leaf_emb (running embedding table)
  hipMemcpyAsync(Wtmp, Leaf, (size_t)VV * DD * sizeof(float),
                 hipMemcpyDeviceToDevice, stream);
  transpose_watt<<<64, 256, 0, stream>>>(W_att, WT);

  for (int lvl = 0; lvl < LVLS; lvl++) {
    hipMemsetAsync(owner, 0xFF, (size_t)VV * sizeof(int), stream);
    attn_kernel<<<NN / 4, 128, 0, stream>>>(lvl, Leaf, Wtmp, nodes, neighbors,
                                            masks, weights, WT, b_att, v_att,
                                            temp);
    owner_kernel<<<(NN + 255) / 256, 256, 0, stream>>>(lvl, nodes, owner);
    scatter_kernel<<<(NN * DD) / 256, 256, 0, stream>>>(lvl, nodes, owner, temp,
                                                        Wtmp);
  }
}